// MaxPool1dDifferentSamplingFreq_73461120631080
// MI455X (gfx1250) — compile-verified
//
#include <hip/hip_runtime.h>
#include <math.h>

// MaxPool1d(K=4, stride=2) over S with ragged NaN tails.
// x: [B=16, C=8, F=64, S=8192] f32 -> out: [16, 8, 64, 4095] f32.
// Memory-bound (~384 MB @ 23.3 TB/s ~= 16.5us). Strategy: async DMA each
// 32KB row into LDS once (ASYNCcnt-tracked global_load_async_to_lds_b128),
// serve the 2x window overlap from LDS banks, coalesced f32 stores.

#define S_LEN 8192
#define P_OUT 4095      // (8192 - 4) / 2 + 1
#define N_ROWS (16 * 8 * 64)

typedef int v4i __attribute__((vector_size(16)));
typedef __attribute__((address_space(1))) v4i* gv4i_p;   // global int4*
typedef __attribute__((address_space(3))) v4i* lv4i_p;   // LDS int4*

__global__ __launch_bounds__(256) void maxpool_ragged_kernel(
    const float* __restrict__ x, float* __restrict__ out) {
  __shared__ float smem[S_LEN];  // 32 KB per block

  const int row = blockIdx.x;
  const int tid = threadIdx.x;
  const float* __restrict__ src = x + (size_t)row * S_LEN;

  // ---- Stage the whole row into LDS -------------------------------------
#if __has_builtin(__builtin_amdgcn_global_load_async_to_lds_b128)
  {
    gv4i_p g = (gv4i_p)(void*)const_cast<float*>(src);
    lv4i_p l = (lv4i_p)(void*)smem;
#pragma unroll
    for (int i = 0; i < 8; ++i) {
      const int v = tid + i * 256;  // 2048 x 16B = 32KB
      __builtin_amdgcn_global_load_async_to_lds_b128(g + v, l + v,
                                                     /*offset=*/0, /*cpol=*/0);
    }
  }
#if __has_builtin(__builtin_amdgcn_s_wait_asynccnt)
  __builtin_amdgcn_s_wait_asynccnt(0);
#else
  asm volatile("s_wait_asynccnt 0" ::: "memory");
#endif
#else
  // Fallback: plain vectorized staging (b128 loads + ds stores).
  {
    const float4* __restrict__ src4 = (const float4*)src;
    float4* sm4 = (float4*)smem;
#pragma unroll
    for (int i = 0; i < 8; ++i) {
      const int v = tid + i * 256;
      sm4[v] = src4[v];
    }
  }
#endif
  __syncthreads();

  // ---- Compute pooled outputs from LDS ----------------------------------
  // Window p covers lds[2p .. 2p+3]. Valid prefix is contiguous per row, so
  // the window is in-bounds of the valid region iff lds[2p+3] is non-NaN.
  // Interleaved p = tid + 256*i: coalesced b32 stores; per-lane LDS reads at
  // word stride 2 across 32 lanes span 64 distinct banks (conflict-free).
  float* __restrict__ dst = out + (size_t)row * P_OUT;
#pragma unroll
  for (int i = 0; i < 16; ++i) {
    const int p = tid + i * 256;
    if (p < P_OUT) {  // only p == 4095 (tid=255, i=15) is culled
      const float a = smem[2 * p + 0];
      const float b = smem[2 * p + 1];
      const float c = smem[2 * p + 2];
      const float d = smem[2 * p + 3];
      const float m = fmaxf(fmaxf(a, b), fmaxf(c, d));
      dst[p] = (d != d) ? __builtin_nanf("") : m;
    }
  }
}

extern "C" void kernel_launch(void* const* d_in, const int* in_sizes, int n_in,
                              void* d_out, int out_size, void* d_ws, size_t ws_size,
                              hipStream_t stream) {
  const float* x = (const float*)d_in[0];
  float* out = (float*)d_out;
  maxpool_ragged_kernel<<<dim3(N_ROWS), dim3(256), 0, stream>>>(x, out);
}